// gatcase30_80814104641734
// MI455X (gfx1250) — compile-verified
//
#include <hip/hip_runtime.h>
#include <math.h>

typedef __attribute__((ext_vector_type(2))) float v2f;
typedef __attribute__((ext_vector_type(8))) float v8f;

#define NUMN   30
#define BATCH  2048
#define NNODE  (BATCH * NUMN)     /* 61440 */
#define NEDGE  (BATCH * 240)      /* 491520 */
#define ETOT   (NEDGE + NNODE)    /* 552960 (edges + self loops) */
#define HEADS  2
#define CH     128
#define HID    256
#define EH     64
#define ROWSO  (BATCH * EH)       /* 131072 */
#define CHUNK  8192

enum { AM_DENSE = 0, AM_CONV = 1, AM_EF = 2 };
enum { EP_NONE = 0, EP_BIAS = 1, EP_BIAS_LRELU = 2 };

// ---------------------------------------------------------------------------
// A-fragment gather helpers (k is even -> float2 never crosses a 256 boundary)
// ---------------------------------------------------------------------------
__device__ inline v2f loadA_conv(const float* __restrict__ A, int r, int k) {
    int tap = k >> 8;          // 0,1,2
    int kk  = k & 255;
    int pos = r % NUMN;
    int sp  = pos + tap - 1;
    v2f z; z.x = 0.f; z.y = 0.f;
    if ((unsigned)sp >= (unsigned)NUMN) return z;
    return *(const v2f*)(A + (size_t)(r + tap - 1) * HID + kk);
}

__device__ inline v2f loadA_ef(const float* __restrict__ A, int r, int k,
                               const int* __restrict__ sidx,
                               const int* __restrict__ didx,
                               const int* __restrict__ vh) {
    int b = r >> 6;            // batch
    int e = r & 63;            // edge within batch
    int node = (k < HID) ? vh[sidx[e]] : vh[didx[e]];
    return *(const v2f*)(A + ((size_t)b * NUMN + node) * HID + (k & 255));
}

// ---------------------------------------------------------------------------
// fp32 WMMA GEMM: C[M,N] = op(A)[M,K] * B[K,N] (+bias)(+leaky)
// wave = 32x32 (2x2 WMMA tiles), block = 8 waves (2x4) = 64x128 tile
// M%64==0, N%128==0, K%4==0 guaranteed by launch sites.
// ---------------------------------------------------------------------------
template <int AMODE, int EPI>
__global__ __launch_bounds__(256) void gemm_wmma_f32(
    const float* __restrict__ A, const float* __restrict__ Bm,
    const float* __restrict__ bias, float* __restrict__ C,
    int M, int Nn, int K, int lda, int row0,
    const int* __restrict__ sidx, const int* __restrict__ didx,
    const int* __restrict__ vh, float slope)
{
    const int tid  = threadIdx.x;
    const int lane = tid & 31;
    const int w    = tid >> 5;
    const int wm   = w >> 2;          // 0..1
    const int wn   = w & 3;           // 0..3
    const int lm   = lane & 15;       // m (A) or n (B) index
    const int hi   = lane >> 4;       // half-wave select
    const int lk   = hi * 2;          // k base within 4-step

    const int tileM = blockIdx.x * 64 + wm * 32;
    const int tileN = blockIdx.y * 128 + wn * 32;

    v8f acc[2][2] = {};
    const int r0 = tileM + lm;
    const int r1 = tileM + 16 + lm;

    for (int k0 = 0; k0 < K; k0 += 4) {
        const int k = k0 + lk;
        v2f a0, a1;
        if (AMODE == AM_DENSE) {
            a0 = *(const v2f*)(A + (size_t)r0 * lda + k);
            a1 = *(const v2f*)(A + (size_t)r1 * lda + k);
        } else if (AMODE == AM_CONV) {
            a0 = loadA_conv(A, r0, k);
            a1 = loadA_conv(A, r1, k);
        } else {
            a0 = loadA_ef(A, row0 + r0, k, sidx, didx, vh);
            a1 = loadA_ef(A, row0 + r1, k, sidx, didx, vh);
        }
        const float* bp = Bm + (size_t)k * Nn + tileN + lm;
        v2f b0, b1;
        b0.x = bp[0];  b0.y = bp[Nn];
        b1.x = bp[16]; b1.y = bp[Nn + 16];
        acc[0][0] = __builtin_amdgcn_wmma_f32_16x16x4_f32(false, a0, false, b0, (short)0, acc[0][0], false, false);
        acc[0][1] = __builtin_amdgcn_wmma_f32_16x16x4_f32(false, a0, false, b1, (short)0, acc[0][1], false, false);
        acc[1][0] = __builtin_amdgcn_wmma_f32_16x16x4_f32(false, a1, false, b0, (short)0, acc[1][0], false, false);
        acc[1][1] = __builtin_amdgcn_wmma_f32_16x16x4_f32(false, a1, false, b1, (short)0, acc[1][1], false, false);
    }

    for (int i = 0; i < 2; ++i)
        for (int j = 0; j < 2; ++j) {
            const int col = tileN + j * 16 + lm;
            float bv = (EPI != EP_NONE) ? bias[col] : 0.f;
#pragma unroll
            for (int t = 0; t < 8; ++t) {
                const int row = tileM + i * 16 + t + 8 * hi;
                float v = acc[i][j][t] + bv;
                if (EPI == EP_BIAS_LRELU) v = v > 0.f ? v : slope * v;
                C[(size_t)row * Nn + col] = v;
            }
        }
}

// ---------------------------------------------------------------------------
// attention dot products: a_src/a_dst per (node, head)
// ---------------------------------------------------------------------------
__global__ void att_dots(const float* __restrict__ xh,
                         const float* __restrict__ att_s, const float* __restrict__ att_d,
                         float* __restrict__ a_s, float* __restrict__ a_d)
{
    int i = blockIdx.x * blockDim.x + threadIdx.x;   // node*2 + head
    if (i >= NNODE * HEADS) return;
    int n = i >> 1, h = i & 1;
    const float* xp = xh + (size_t)n * HID + h * CH;
    float ss = 0.f, sd = 0.f;
    for (int c = 0; c < CH; ++c) { float v = xp[c]; ss += v * att_s[h * CH + c]; sd += v * att_d[h * CH + c]; }
    a_s[i] = ss; a_d[i] = sd;
}

// out rows init with gat bias; amax=-inf, denom=0
__global__ void gat_init(float* __restrict__ out, const float* __restrict__ gat_b,
                         float* __restrict__ amax, float* __restrict__ den)
{
    size_t i = (size_t)blockIdx.x * 256 + threadIdx.x;
    if (i < (size_t)NNODE * HID) out[i] = gat_b[i & (HID - 1)];
    if (i < (size_t)NNODE * HEADS) { amax[i] = -INFINITY; den[i] = 0.f; }
}

__device__ inline void atomic_max_f32(float* addr, float val) {
    if (val >= 0.f) atomicMax((int*)addr, __float_as_int(val));
    else            atomicMin((unsigned int*)addr, __float_as_uint(val));
}

__device__ inline void edge_sd(int e, const int* __restrict__ ei, int& s, int& d) {
    if (e < NEDGE) { s = ei[e]; d = ei[NEDGE + e]; }
    else           { s = d = e - NEDGE; }
}

__global__ void edge_amax(const int* __restrict__ ei, const float* __restrict__ a_s,
                          const float* __restrict__ a_d, float* __restrict__ amax)
{
    int e = blockIdx.x * 256 + threadIdx.x;
    if (e >= ETOT) return;
    int s, d; edge_sd(e, ei, s, d);
    for (int h = 0; h < HEADS; ++h) {
        float v = a_s[s * 2 + h] + a_d[d * 2 + h];
        v = v > 0.f ? v : 0.2f * v;
        atomic_max_f32(&amax[d * 2 + h], v);
    }
}

__global__ void edge_denom(const int* __restrict__ ei, const float* __restrict__ a_s,
                           const float* __restrict__ a_d, const float* __restrict__ amax,
                           float* __restrict__ den)
{
    int e = blockIdx.x * 256 + threadIdx.x;
    if (e >= ETOT) return;
    int s, d; edge_sd(e, ei, s, d);
    for (int h = 0; h < HEADS; ++h) {
        float v = a_s[s * 2 + h] + a_d[d * 2 + h];
        v = v > 0.f ? v : 0.2f * v;
        atomicAdd(&den[d * 2 + h], expf(v - amax[d * 2 + h]));
    }
}

// one block of 256 threads handles 4 edges; 64 threads/edge, 4 channels/thread
__global__ __launch_bounds__(256) void edge_scatter(
    const int* __restrict__ ei, const float* __restrict__ a_s,
    const float* __restrict__ a_d, const float* __restrict__ amax,
    const float* __restrict__ den, const float* __restrict__ xh,
    float* __restrict__ out)
{
    int e = blockIdx.x * 4 + (threadIdx.x >> 6);
    if (e >= ETOT) return;
    int t = threadIdx.x & 63;
    int s, d; edge_sd(e, ei, s, d);
    int c = t * 4;                 // channel base, 0..252
    int h = c >> 7;                // head
    float v = a_s[s * 2 + h] + a_d[d * 2 + h];
    v = v > 0.f ? v : 0.2f * v;
    float al = expf(v - amax[d * 2 + h]);
    float wgt = al / (den[d * 2 + h] + 1e-16f);
    const float4 xv = *(const float4*)(xh + (size_t)s * HID + c);
    float* op = out + (size_t)d * HID + c;
    atomicAdd(op + 0, xv.x * wgt);
    atomicAdd(op + 1, xv.y * wgt);
    atomicAdd(op + 2, xv.z * wgt);
    atomicAdd(op + 3, xv.w * wgt);
}

// ---------------------------------------------------------------------------
// layer norm + activation (MODE 0: leaky 0.01; MODE 1: resid + relu)
// ---------------------------------------------------------------------------
template <int MODE>
__global__ __launch_bounds__(256) void ln_act(
    const float* __restrict__ in, const float* __restrict__ g,
    const float* __restrict__ b, const float* __restrict__ resid,
    float* __restrict__ out)
{
    __shared__ float sb[HID];
    const int r = blockIdx.x, tid = threadIdx.x;
    float x = in[(size_t)r * HID + tid];
    sb[tid] = x; __syncthreads();
    for (int s = 128; s > 0; s >>= 1) { if (tid < s) sb[tid] += sb[tid + s]; __syncthreads(); }
    float mu = sb[0] * (1.f / HID);
    __syncthreads();
    float dx = x - mu;
    sb[tid] = dx * dx; __syncthreads();
    for (int s = 128; s > 0; s >>= 1) { if (tid < s) sb[tid] += sb[tid + s]; __syncthreads(); }
    float var = sb[0] * (1.f / HID);
    float y = dx * rsqrtf(var + 1e-5f) * g[tid] + b[tid];
    if (MODE == 0) { y = y > 0.f ? y : 0.01f * y; out[(size_t)r * HID + tid] = y; }
    else { y = y > 0.f ? y : 0.f; out[(size_t)r * HID + tid] = resid[(size_t)r * HID + tid] + y; }
}

// conv_w[o][i][t] (O,I,3 row-major) -> B[k=t*256+i][n=o]
__global__ void pack_convw(const float* __restrict__ cw, float* __restrict__ wc) {
    int i = blockIdx.x * 256 + threadIdx.x;   // k*256 + n
    if (i >= 768 * HID) return;
    int k = i >> 8, n = i & 255;
    wc[i] = cw[n * 768 + (k & 255) * 3 + (k >> 8)];
}

__global__ void fc4_sigmoid(const float* __restrict__ in, const float* __restrict__ w,
                            const float* __restrict__ b, float* __restrict__ out,
                            int rows, int row0)
{
    int r = blockIdx.x * 256 + threadIdx.x;
    if (r >= rows) return;
    const float* p = in + (size_t)r * HID;
    float s = 0.f;
    for (int c = 0; c < HID; ++c) s += p[c] * w[c];
    s += b[0];
    out[row0 + r] = 1.f / (1.f + expf(-s));
}

// ---------------------------------------------------------------------------
// host launch
// ---------------------------------------------------------------------------
extern "C" void kernel_launch(void* const* d_in, const int* in_sizes, int n_in,
                              void* d_out, int out_size, void* d_ws, size_t ws_size,
                              hipStream_t stream)
{
    (void)in_sizes; (void)n_in; (void)out_size; (void)ws_size;

    // input leaves (jax pytree order: top dict insertion order; params dict keys sorted)
    const float* x    = (const float*)d_in[0];   // N x 64
    const int*   ei   = (const int*)d_in[1];     // 2 x E
    const int*   sidx = (const int*)d_in[2];     // EH
    const int*   didx = (const int*)d_in[3];     // EH
    const int*   vh   = (const int*)d_in[4];     // 30
    const float* fc1_b = (const float*)d_in[5];
    const float* fc1_w = (const float*)d_in[6];
    const float* fc2_b = (const float*)d_in[7];
    const float* fc2_w = (const float*)d_in[8];
    const float* fc3_b = (const float*)d_in[9];
    const float* fc3_w = (const float*)d_in[10];
    const float* fc4_b = (const float*)d_in[11];
    const float* fc4_w = (const float*)d_in[12];
    // layers[l]: W, att_dst, att_src, conv_b, conv_w, gat_b, ln1_b, ln1_g, ln2_b, ln2_g
    auto L = [&](int l, int j) { return (const float*)d_in[13 + 10 * l + j]; };

    // workspace layout (floats)
    float* ws   = (float*)d_ws;
    float* h    = ws;                                  // N*HID
    float* xh   = h   + (size_t)NNODE * HID;           // N*HID (also reused as hc)
    float* gat  = xh  + (size_t)NNODE * HID;           // N*HID
    float* a_s  = gat + (size_t)NNODE * HID;           // N*2
    float* a_d  = a_s + (size_t)NNODE * HEADS;
    float* amax = a_d + (size_t)NNODE * HEADS;
    float* den  = amax + (size_t)NNODE * HEADS;
    float* wc   = den  + (size_t)NNODE * HEADS;        // 768*HID
    float* o1   = wc   + (size_t)768 * HID;            // CHUNK*1024
    float* o2   = o1   + (size_t)CHUNK * 1024;         // CHUNK*256
    float* o3   = o2   + (size_t)CHUNK * 256;          // CHUNK*256
    float* outp = (float*)d_out;

    const dim3 blk(256);

    for (int l = 0; l < 3; ++l) {
        const float* hin = (l == 0) ? x : h;
        const int Kin = (l == 0) ? 64 : HID;

        // xh = hin @ W
        gemm_wmma_f32<AM_DENSE, EP_NONE><<<dim3(NNODE / 64, HID / 128), blk, 0, stream>>>(
            hin, L(l, 0), nullptr, xh, NNODE, HID, Kin, Kin, 0, nullptr, nullptr, nullptr, 0.f);

        // attention dots
        att_dots<<<dim3((NNODE * HEADS + 255) / 256), blk, 0, stream>>>(xh, L(l, 2), L(l, 1), a_s, a_d);

        // init out rows with bias, amax/-inf, denom/0
        gat_init<<<dim3((NNODE * HID) / 256), blk, 0, stream>>>(gat, L(l, 5), amax, den);

        // segment softmax + scatter aggregate
        edge_amax<<<dim3((ETOT + 255) / 256), blk, 0, stream>>>(ei, a_s, a_d, amax);
        edge_denom<<<dim3((ETOT + 255) / 256), blk, 0, stream>>>(ei, a_s, a_d, amax, den);
        edge_scatter<<<dim3((ETOT + 3) / 4), blk, 0, stream>>>(ei, a_s, a_d, amax, den, xh, gat);

        // h_gat = leaky(LN1(gat)) in place
        ln_act<0><<<dim3(NNODE), blk, 0, stream>>>(gat, L(l, 7), L(l, 6), nullptr, gat);

        // conv as K=768 GEMM with on-the-fly 3-tap im2col; hc -> xh (reused)
        pack_convw<<<dim3((768 * HID) / 256), blk, 0, stream>>>(L(l, 4), wc);
        gemm_wmma_f32<AM_CONV, EP_BIAS><<<dim3(NNODE / 64, HID / 128), blk, 0, stream>>>(
            gat, wc, L(l, 3), xh, NNODE, HID, 768, HID, 0, nullptr, nullptr, nullptr, 0.f);

        // h = h_gat + relu(LN2(hc))
        ln_act<1><<<dim3(NNODE), blk, 0, stream>>>(xh, L(l, 9), L(l, 8), gat, h);
    }

    // edge-pair MLP, chunked over rows; fc1 fuses the [h_src|h_dst] concat gather
    for (int row0 = 0; row0 < ROWSO; row0 += CHUNK) {
        gemm_wmma_f32<AM_EF, EP_BIAS_LRELU><<<dim3(CHUNK / 64, 1024 / 128), blk, 0, stream>>>(
            h, fc1_w, fc1_b, o1, CHUNK, 1024, 2 * HID, HID, row0, sidx, didx, vh, 0.01f);
        gemm_wmma_f32<AM_DENSE, EP_BIAS_LRELU><<<dim3(CHUNK / 64, 256 / 128), blk, 0, stream>>>(
            o1, fc2_w, fc2_b, o2, CHUNK, 256, 1024, 1024, 0, nullptr, nullptr, nullptr, 0.01f);
        gemm_wmma_f32<AM_DENSE, EP_BIAS_LRELU><<<dim3(CHUNK / 64, 256 / 128), blk, 0, stream>>>(
            o2, fc3_w, fc3_b, o3, CHUNK, 256, 256, 256, 0, nullptr, nullptr, nullptr, 0.01f);
        fc4_sigmoid<<<dim3(CHUNK / 256), blk, 0, stream>>>(o3, fc4_w, fc4_b, outp, CHUNK, row0);
    }
}